// GCNBranch_12807592476805
// MI455X (gfx1250) — compile-verified
//
#include <hip/hip_runtime.h>

#define NN 50000
#define NE 800000
#define D  128

typedef __attribute__((ext_vector_type(2))) float v2f;
typedef __attribute__((ext_vector_type(8))) float v8f;

// ---------------------------------------------------------------- degrees
__global__ void degree_kernel(const int* __restrict__ src,
                              const int* __restrict__ dst,
                              float* __restrict__ din,
                              float* __restrict__ dout, int E) {
    int e = blockIdx.x * blockDim.x + threadIdx.x;
    if (e < E) {
        unsafeAtomicAdd(&din[dst[e]], 1.0f);
        unsafeAtomicAdd(&dout[src[e]], 1.0f);
    }
}

// ------------------------------------------------- edge gather + scatter-add
// one wave per edge; lane l moves features [4l, 4l+4)
__global__ __launch_bounds__(256)
void scatter_kernel(const float* __restrict__ h,
                    const int* __restrict__ src,
                    const int* __restrict__ dst,
                    const float* __restrict__ degout,
                    float* __restrict__ agg, int E, int useScale) {
    int wave = (blockIdx.x * blockDim.x + threadIdx.x) >> 5;
    int lane = threadIdx.x & 31;
    if (wave >= E) return;
    int s = src[wave];
    int d = dst[wave];
    float sc = 1.0f;
    if (useScale) sc = rsqrtf(fmaxf(degout[s], 1.0f));   // D_out^{-1/2}, clamped
    const float4* hv = reinterpret_cast<const float4*>(h + (size_t)s * D);
    float4 v = hv[lane];
    float* ap = agg + (size_t)d * D + lane * 4;
    unsafeAtomicAdd(ap + 0, v.x * sc);
    unsafeAtomicAdd(ap + 1, v.y * sc);
    unsafeAtomicAdd(ap + 2, v.z * sc);
    unsafeAtomicAdd(ap + 3, v.w * sc);
}

// ------------------------------------- fused row-normalize + GEMM (f32 WMMA)
// mode 0 (SAGE-gcn):  A[r,:] = (agg[r,:] + hprev[r,:]) / (deg[r] + 1)
// mode 1 (GraphConv): A[r,:] =  agg[r,:] * rsqrt(max(deg[r],1))
// out = A @ W + bias ; one block = 16 rows x 128 cols, 8 waves x 16-col tiles
__global__ __launch_bounds__(256)
void norm_gemm_wmma(const float* __restrict__ agg,
                    const float* __restrict__ hprev,
                    const float* __restrict__ deg,
                    const float* __restrict__ W,
                    const float* __restrict__ bias,
                    float* __restrict__ out, int mode) {
    __shared__ float Alds[16 * 132];            // stride 132: conflict-free frags

    const int t = threadIdx.x;
    const int rowBase = blockIdx.x * 16;        // 50000 = 16*3125 exactly

    // cooperative normalized A-tile load: thread -> (row = t/16, 8 cols)
    {
        int r  = t >> 4;
        int c0 = (t & 15) * 8;
        int row = rowBase + r;
        float s;
        if (mode == 0) s = 1.0f / (deg[row] + 1.0f);
        else           s = rsqrtf(fmaxf(deg[row], 1.0f));
        const float* ar = agg   + (size_t)row * D + c0;
        const float* hr = hprev + (size_t)row * D + c0;
#pragma unroll
        for (int j = 0; j < 8; ++j) {
            float v = ar[j];
            if (mode == 0) v += hr[j];
            Alds[r * 132 + c0 + j] = v * s;
        }
    }
    __syncthreads();

    const int waveId = t >> 5;                  // 0..7 -> 16-col panel
    const int lane   = t & 31;
    const int half   = lane >> 4;               // lanes 16-31 hold K+2 / M+8
    const int lm     = lane & 15;
    const int col0   = waveId * 16;

    v8f c = {};
#pragma unroll
    for (int kk = 0; kk < 32; ++kk) {           // K = 128 = 32 * 4
        int k0 = 4 * kk + 2 * half;
        v2f a, b;
        // A 16x4 f32 layout: lanes0-15 = {K0,K1}, lanes16-31 = {K2,K3}
        a.x = Alds[lm * 132 + k0];
        a.y = Alds[lm * 132 + k0 + 1];
        // B 4x16 f32: row k striped across lanes within a VGPR
        b.x = W[(size_t)(k0)     * D + col0 + lm];
        b.y = W[(size_t)(k0 + 1) * D + col0 + lm];
        c = __builtin_amdgcn_wmma_f32_16x16x4_f32(
                false, a, false, b, (short)0, c, false, false);
    }

    float bb = bias[col0 + lm];
#pragma unroll
    for (int v = 0; v < 8; ++v) {               // C/D: M = v + 8*half
        int row = rowBase + v + 8 * half;
        out[(size_t)row * D + col0 + lm] = c[v] + bb;
    }
}

// ---------------------------------------------------------------- launcher
extern "C" void kernel_launch(void* const* d_in, const int* in_sizes, int n_in,
                              void* d_out, int out_size, void* d_ws, size_t ws_size,
                              hipStream_t stream) {
    const float* x  = (const float*)d_in[0];
    const float* W1 = (const float*)d_in[1];
    const float* b1 = (const float*)d_in[2];
    const float* W2 = (const float*)d_in[3];
    const float* b2 = (const float*)d_in[4];
    const float* W3 = (const float*)d_in[5];
    const float* b3 = (const float*)d_in[6];
    const int* src  = (const int*)d_in[7];
    const int* dst  = (const int*)d_in[8];
    float* out = (float*)d_out;

    const int N = NN, E = NE;
    float* deg_in  = (float*)d_ws;
    float* deg_out = deg_in + N;
    float* agg     = deg_out + N;
    float* h1      = agg + (size_t)N * D;
    float* h2      = h1  + (size_t)N * D;

    const size_t matBytes = (size_t)N * D * sizeof(float);
    dim3 edgeWaveGrid((E + 7) / 8);     // 8 edges (waves) per 256-thread block
    dim3 gemmGrid(N / 16);

    hipMemsetAsync(deg_in, 0, (size_t)2 * N * sizeof(float), stream);
    hipMemsetAsync(agg, 0, matBytes, stream);
    degree_kernel<<<(E + 255) / 256, 256, 0, stream>>>(src, dst, deg_in, deg_out, E);

    // layer 1: SAGE-gcn
    scatter_kernel<<<edgeWaveGrid, 256, 0, stream>>>(x, src, dst, deg_out, agg, E, 0);
    norm_gemm_wmma<<<gemmGrid, 256, 0, stream>>>(agg, x, deg_in, W1, b1, h1, 0);

    // layer 2: SAGE-gcn
    hipMemsetAsync(agg, 0, matBytes, stream);
    scatter_kernel<<<edgeWaveGrid, 256, 0, stream>>>(h1, src, dst, deg_out, agg, E, 0);
    norm_gemm_wmma<<<gemmGrid, 256, 0, stream>>>(agg, h1, deg_in, W2, b2, h2, 0);

    // layer 3: GraphConv norm='both' (scale sources by D_out^-1/2, dests by D_in^-1/2)
    hipMemsetAsync(agg, 0, matBytes, stream);
    scatter_kernel<<<edgeWaveGrid, 256, 0, stream>>>(h2, src, dst, deg_out, agg, E, 1);
    norm_gemm_wmma<<<gemmGrid, 256, 0, stream>>>(agg, h2, deg_in, W3, b3, out, 1);
}